// SWAChunk1_386547056965
// MI455X (gfx1250) — compile-verified
//
#include <hip/hip_runtime.h>
#include <math.h>

#define HDIM 2048
#define NH 8
#define NKV 2
#define HD 256
#define WIN 512
#define CTX 4096
#define DFF 2048
#define PLD 256
#define NTOT 35
#define LAYERS 8
#define POSI 1024
#define EPSV 1e-6f
#define MAXK 1056   // ceil(1025/16)*16 = 1040, padded to 66*16

typedef __attribute__((ext_vector_type(2))) float v2f;
typedef __attribute__((ext_vector_type(8))) float v8f;
typedef __attribute__((ext_vector_type(4))) unsigned int v4u;
typedef __attribute__((ext_vector_type(8))) int v8i;
typedef __attribute__((ext_vector_type(4))) int v4i;

__device__ __forceinline__ float gelu_tanh(float x) {
  float x3 = x * x * x;
  return 0.5f * x * (1.f + tanhf(0.7978845608028654f * (x + 0.044715f * x3)));
}

// ---------------- trivial utility kernels ----------------
__global__ void zero_k(float* p, int n) {
  int i = blockIdx.x * blockDim.x + threadIdx.x;
  if (i < n) p[i] = 0.f;
}
__global__ void copy_k(float* dst, const float* src, int n) {
  int i = blockIdx.x * blockDim.x + threadIdx.x;
  if (i < n) dst[i] = src[i];
}

// ---------------- RMS norm: out = rms(in)*w, one block ----------------
__global__ void rms_k(float* out, const float* in, const float* w, int len) {
  __shared__ float rbuf[256];
  int t = threadIdx.x;
  float s = 0.f;
  for (int i = t; i < len; i += 256) { float v = in[i]; s += v * v; }
  rbuf[t] = s; __syncthreads();
  for (int o = 128; o; o >>= 1) { if (t < o) rbuf[t] += rbuf[t + o]; __syncthreads(); }
  float r = rsqrtf(rbuf[0] / (float)len + EPSV);
  for (int i = t; i < len; i += 256) out[i] = in[i] * r * (w ? w[i] : 1.f);
}

// ------------- h = (h + rms(y)*w) * (ls ? ls[li] : 1), one block -------------
__global__ void addrms_k(float* h, const float* y, const float* w, int len,
                         const float* ls, int li) {
  __shared__ float rbuf[256];
  int t = threadIdx.x;
  float s = 0.f;
  for (int i = t; i < len; i += 256) { float v = y[i]; s += v * v; }
  rbuf[t] = s; __syncthreads();
  for (int o = 128; o; o >>= 1) { if (t < o) rbuf[t] += rbuf[t + o]; __syncthreads(); }
  float r = rsqrtf(rbuf[0] / (float)len + EPSV);
  float sc = ls ? ls[li] : 1.f;
  for (int i = t; i < len; i += 256) h[i] = (h[i] + y[i] * r * w[i]) * sc;
}

// ---------------- GEMV: out[n] += sum_k x[k]*W[k*N+n], split-K atomics ----------------
// grid.x = col tiles (1024 cols each), grid.y = K / kchunk, block = 256
__global__ void gemv_k(const float* __restrict__ x, const float* __restrict__ W,
                       float* __restrict__ out, int K, int N, int kchunk) {
  __shared__ float xs[256];
  int k0 = blockIdx.y * kchunk;
  for (int i = threadIdx.x; i < kchunk; i += 256) xs[i] = x[k0 + i];
  __syncthreads();
  int col = (blockIdx.x * 256 + threadIdx.x) * 4;
  if (col >= N) return;
  const float* Wp = W + (size_t)k0 * N + col;
  float ax = 0.f, ay = 0.f, az = 0.f, aw = 0.f;
  for (int kk = 0; kk < kchunk; kk++) {
    float4 wv = *(const float4*)Wp;
    __builtin_prefetch(Wp + (size_t)32 * N, 0, 0);   // global_prefetch_b8
    float xv = xs[kk];
    ax = fmaf(xv, wv.x, ax); ay = fmaf(xv, wv.y, ay);
    az = fmaf(xv, wv.z, az); aw = fmaf(xv, wv.w, aw);
    Wp += N;
  }
  atomicAdd(out + col + 0, ax);
  atomicAdd(out + col + 1, ay);
  atomicAdd(out + col + 2, az);
  atomicAdd(out + col + 3, aw);
}

// --------- per-head RMS (+opt weight) then RoPE, in place. grid = NH+2*NKV, block=HD ---------
__global__ void qkvnorm_k(float* q, float* k, float* v, const float* qn,
                          const float* kn, const float* cosv, const float* sinv) {
  int b = blockIdx.x, t = threadIdx.x;
  float* vec; const float* w; int rope;
  if (b < NH)            { vec = q + b * HD;              w = qn;      rope = 1; }
  else if (b < NH + NKV) { vec = k + (b - NH) * HD;       w = kn;      rope = 1; }
  else                   { vec = v + (b - NH - NKV) * HD; w = nullptr; rope = 0; }
  __shared__ float buf[HD];
  __shared__ float rbuf[HD];
  float x = vec[t];
  rbuf[t] = x * x; __syncthreads();
  for (int o = HD / 2; o; o >>= 1) { if (t < o) rbuf[t] += rbuf[t + o]; __syncthreads(); }
  float r = rsqrtf(rbuf[0] / (float)HD + EPSV);
  float n = x * r * (w ? w[t] : 1.f);
  buf[t] = n; __syncthreads();
  float o;
  if (rope) {
    float rot = (t < HD / 2) ? -buf[t + HD / 2] : buf[t - HD / 2];
    o = n * cosv[t] + rot * sinv[t];
  } else o = n;
  vec[t] = o;
}

// ---------------- sliding cache: shift left 1, append new token ----------------
__global__ void slide_k(float* Kout, float* Vout, const float* Kin, const float* Vin,
                        const float* knew, const float* vnew) {
  int idx = blockIdx.x * blockDim.x + threadIdx.x;       // NKV*WIN*HD
  if (idx >= NKV * WIN * HD) return;
  int d = idx % HD; int rest = idx / HD;
  int wpos = rest % WIN; int kv = rest / WIN;
  if (wpos < WIN - 1) {
    int src = (kv * WIN + wpos + 1) * HD + d;
    Kout[idx] = Kin[src]; Vout[idx] = Vin[src];
  } else {
    Kout[idx] = knew[kv * HD + d]; Vout[idx] = vnew[kv * HD + d];
  }
}

// ---------------- full cache: copy, replace row POSI ----------------
__global__ void fullupd_k(float* Kout, float* Vout, const float* Kin, const float* Vin,
                          const float* knew, const float* vnew) {
  int idx = blockIdx.x * blockDim.x + threadIdx.x;       // NKV*CTX*HD
  if (idx >= NKV * CTX * HD) return;
  int d = idx % HD; int rest = idx / HD;
  int p = rest % CTX; int kv = rest / CTX;
  if (p == POSI) { Kout[idx] = knew[kv * HD + d]; Vout[idx] = vnew[kv * HD + d]; }
  else           { Kout[idx] = Kin[idx];          Vout[idx] = Vin[idx]; }
}

// ---------------- WMMA attention: one block per KV group, 8 waves ----------------
// scores[m][key] = sum_d q[m][d]*K[key][d]   (M rows 0..3 = heads, N=16 keys, K-chunk 4)
// out[m][d]     = sum_k P[m][k]*V[k][d]      (M rows 0..3 = heads, N=16 d-cols, K-chunk 4)
// A rows 4..15 are duplicates of rows 0..3 (via m&3): D rows 4..15 are never read,
// and D[m][*] depends only on A[m][*], so this is exact and keeps EXEC uniform.
__global__ void attn_k(const float* __restrict__ q, const float* __restrict__ Kc,
                       const float* __restrict__ Vc, float* __restrict__ ao,
                       int Lbuf, int nkeys) {
  int g = blockIdx.x;
  int tid = threadIdx.x, lane = tid & 31, wave = tid >> 5;
  __shared__ float qs[4 * HD];
  __shared__ float S[4 * MAXK];
  __shared__ float rbuf[256];

#if __has_builtin(__builtin_amdgcn_tensor_load_to_lds) && __has_builtin(__builtin_amdgcn_s_wait_tensorcnt)
  // Tensor Data Mover: async-stage the 4KB q block (1-D tile, 1024 x 4B) into LDS.
  // The cooperative copy below writes the identical bytes, so the result is
  // deterministic regardless of TDM timing.
  if (tid < 32) {
    unsigned long long ga = (unsigned long long)(const void*)(q + g * 4 * HD);
    unsigned int lds = (unsigned int)(unsigned long long)(void*)qs;  // LDS byte offset
    unsigned int nel = 4 * HD;
    v4u g0;
    g0.x = 1u;                                           // count=1 (valid user D#)
    g0.y = lds;                                          // lds_addr
    g0.z = (unsigned int)(ga & 0xffffffffu);             // global_addr[31:0]
    g0.w = (unsigned int)((ga >> 32) & 0x01ffffffu) | (2u << 30); // addr[56:32], type=2
    v8i g1;
    g1[0] = (int)(2u << 16);                 // workgroup_mask=0, data_size=2 (4B)
    g1[1] = (int)((nel & 0xffffu) << 16);    // tensor_dim0[15:0]
    g1[2] = (int)((nel >> 16) & 0xffffu);    // tensor_dim0[31:16], tensor_dim1[15:0]=0
    g1[3] = (int)((nel & 0xffffu) << 16);    // tile_dim0 = nel
    g1[4] = 0;                               // tile_dim1 = tile_dim2 = 0 (1-D)
    g1[5] = (int)nel;                        // tensor_dim0_stride[31:0]
    g1[6] = 0;
    g1[7] = 0;
    v4i gz = {0, 0, 0, 0};
    v8i gz8 = {0, 0, 0, 0, 0, 0, 0, 0};
    __builtin_amdgcn_tensor_load_to_lds(g0, g1, gz, gz, gz8, 0);
    __builtin_amdgcn_s_wait_tensorcnt(0);
  }
#endif
  for (int i = tid; i < 4 * HD; i += 256) qs[i] = q[g * 4 * HD + i];
  for (int i = tid; i < 4 * MAXK; i += 256) S[i] = -3.0e4f;
  __syncthreads();

  const float* Kg = Kc + (size_t)g * Lbuf * HD;
  const float* Vg = Vc + (size_t)g * Lbuf * HD;
  int nkt = (nkeys + 15) / 16;
  int kpad = nkt * 16;
  int m = lane & 15;
  int mh = m & 3;
  int dbase = (lane < 16) ? 0 : 2;  // ISA A/B layout: lanes 0-15 carry K=0/1, 16-31 carry K=2/3

  // ---- scores via v_wmma_f32_16x16x4_f32 ----
  for (int t = wave; t < nkt; t += 8) {
    int kb = t * 16;
    int key = kb + m;                       // B layout: N = lane&15 in both half-waves
    const float* kp = Kg + (size_t)key * HD + dbase;
    const float* ap = qs + mh * HD + dbase;
    v8f c = {};
    for (int d0 = 0; d0 < HD; d0 += 4) {
      v2f a = *(const v2f*)(ap + d0);       // ds_load_b64, uniform EXEC
      v2f b = *(const v2f*)(kp + d0);       // global_load_b64
      c = __builtin_amdgcn_wmma_f32_16x16x4_f32(false, a, false, b,
                                                (short)0, c, false, false);
    }
    // D layout: VGPR r -> M=r on lanes 0-15. Heads are rows 0..3.
    if (lane < 16 && key < nkeys) {
      S[0 * MAXK + key] = c[0];
      S[1 * MAXK + key] = c[1];
      S[2 * MAXK + key] = c[2];
      S[3 * MAXK + key] = c[3];
    }
  }
  __syncthreads();

  // ---- softmax per head (mask is 0 on valid keys; invalid keys dropped exactly) ----
  for (int h = 0; h < 4; h++) {
    float mx = -3.0e4f;
    for (int i = tid; i < nkeys; i += 256) mx = fmaxf(mx, S[h * MAXK + i]);
    rbuf[tid] = mx; __syncthreads();
    for (int o = 128; o; o >>= 1) { if (tid < o) rbuf[tid] = fmaxf(rbuf[tid], rbuf[tid + o]); __syncthreads(); }
    float mm = rbuf[0]; __syncthreads();
    float s = 0.f;
    for (int i = tid; i < kpad; i += 256) {
      float e = (i < nkeys) ? __expf(S[h * MAXK + i] - mm) : 0.f;
      S[h * MAXK + i] = e; s += e;
    }
    rbuf[tid] = s; __syncthreads();
    for (int o = 128; o; o >>= 1) { if (tid < o) rbuf[tid] += rbuf[tid + o]; __syncthreads(); }
    float inv = 1.f / rbuf[0]; __syncthreads();
    for (int i = tid; i < kpad; i += 256) S[h * MAXK + i] *= inv;
    __syncthreads();
  }

  // ---- P @ V via v_wmma_f32_16x16x4_f32 ----
  for (int dt = wave; dt < HD / 16; dt += 8) {
    int db = dt * 16;
    int dcol = db + m;
    const float* Sp = S + mh * MAXK + dbase;
    v8f c = {};
    for (int k0 = 0; k0 < kpad; k0 += 4) {
      int krow = k0 + dbase;
      v2f a = *(const v2f*)(Sp + k0);       // ds_load_b64, uniform EXEC
      v2f b;
      b.x = Vg[(size_t)krow * HD + dcol];
      b.y = Vg[(size_t)(krow + 1) * HD + dcol];
      c = __builtin_amdgcn_wmma_f32_16x16x4_f32(false, a, false, b,
                                                (short)0, c, false, false);
    }
    if (lane < 16) {
      for (int h = 0; h < 4; h++)
        ao[(g * 4 + h) * HD + db + lane] = c[h];
    }
  }
}

// ---------------- PLE combine: rms(proj*ps)*nw + raw, times is. grid=NTOT, block=PLD ----------------
__global__ void ple_k(float* comb, const float* proj, const float* raw,
                      const float* nw, const float* ps, const float* is) {
  int c = blockIdx.x, t = threadIdx.x;
  __shared__ float rbuf[PLD];
  float v = proj[c * PLD + t] * ps[0];
  rbuf[t] = v * v; __syncthreads();
  for (int o = PLD / 2; o; o >>= 1) { if (t < o) rbuf[t] += rbuf[t + o]; __syncthreads(); }
  float r = rsqrtf(rbuf[0] / (float)PLD + EPSV);
  comb[c * PLD + t] = (v * r * nw[t] + raw[c * PLD + t]) * is[0];
}

__global__ void glu_k(float* ff, const float* g, const float* u, int n) {
  int i = blockIdx.x * blockDim.x + threadIdx.x;
  if (i < n) ff[i] = gelu_tanh(g[i]) * u[i];
}
__global__ void plgate_k(float* pg, const float* comb, int n) {
  int i = blockIdx.x * blockDim.x + threadIdx.x;
  if (i < n) pg[i] = gelu_tanh(pg[i]) * comb[i];
}

// =====================================================================
extern "C" void kernel_launch(void* const* d_in, const int* in_sizes, int n_in,
                              void* d_out, int out_size, void* d_ws, size_t ws_size,
                              hipStream_t stream) {
  (void)in_sizes; (void)n_in; (void)out_size; (void)ws_size;
  const float* hidden = (const float*)d_in[0];
  const float* plraw  = (const float*)d_in[4];
  const float* cos_s  = (const float*)d_in[5];
  const float* sin_s  = (const float*)d_in[6];
  const float* cos_f  = (const float*)d_in[7];
  const float* sin_f  = (const float*)d_in[8];
  const float* Ksin   = (const float*)d_in[9];
  const float* Vsin   = (const float*)d_in[10];
  const float* Kfin   = (const float*)d_in[11];
  const float* Vfin   = (const float*)d_in[12];
  const float* ln_in  = (const float*)d_in[13];
  const float* ln_pa  = (const float*)d_in[14];
  const float* ln_pf  = (const float*)d_in[15];
  const float* ln_ff  = (const float*)d_in[16];
  const float* ln_pl  = (const float*)d_in[17];
  const float* Wq     = (const float*)d_in[18];
  const float* Wk     = (const float*)d_in[19];
  const float* Wv     = (const float*)d_in[20];
  const float* Wo     = (const float*)d_in[21];
  const float* qn     = (const float*)d_in[22];
  const float* kn     = (const float*)d_in[23];
  const float* Wgate  = (const float*)d_in[24];
  const float* Wup    = (const float*)d_in[25];
  const float* Wdown  = (const float*)d_in[26];
  const float* Wplg   = (const float*)d_in[27];
  const float* Wplp   = (const float*)d_in[28];
  const float* lsc    = (const float*)d_in[29];
  const float* Wple   = (const float*)d_in[30];
  const float* plnw   = (const float*)d_in[31];
  const float* ppsc   = (const float*)d_in[32];
  const float* pisc   = (const float*)d_in[33];

  float* out = (float*)d_out;
  float* Ks = out + HDIM;                       // 7*NKV*WIN*HD
  float* Vs = Ks + 7 * NKV * WIN * HD;
  float* Kf = Vs + 7 * NKV * WIN * HD;          // NKV*CTX*HD
  float* Vf = Kf + NKV * CTX * HD;

  float* ws    = (float*)d_ws;
  float* wh    = ws;            // 2048 running hidden
  float* wx    = ws + 2048;     // 2048 normed
  float* wq    = ws + 4096;     // 2048
  float* wkv   = ws + 6144;     // k 512
  float* wvv   = ws + 6656;     // v 512
  float* wao   = ws + 7168;     // 2048
  float* wy    = ws + 9216;     // 2048
  float* wg    = ws + 11264;    // 2048
  float* wu    = ws + 13312;    // 2048
  float* wff   = ws + 15360;    // 2048
  float* wpg   = ws + 17408;    // 256
  float* wproj = ws + 17664;    // 8960
  float* wcomb = ws + 26624;    // 8960

  auto zero = [&](float* p, int n) {
    zero_k<<<(n + 255) / 256, 256, 0, stream>>>(p, n);
  };
  auto gemv = [&](const float* x, const float* W, float* o, int K, int N) {
    zero(o, N);
    int kchunk = (K >= 2048) ? 128 : 64;
    dim3 grid((N / 4 + 255) / 256, K / kchunk);
    gemv_k<<<grid, 256, 0, stream>>>(x, W, o, K, N, kchunk);
  };

  static const int SI[LAYERS] = {0, 1, 2, 3, -1, 4, 5, 6};

  // h = hidden_states
  copy_k<<<8, 256, 0, stream>>>(wh, hidden, HDIM);

  // PLE: proj = hidden @ Wple, then per-chunk rms + raw
  gemv(hidden, Wple, wproj, HDIM, NTOT * PLD);
  ple_k<<<NTOT, PLD, 0, stream>>>(wcomb, wproj, plraw, plnw, ppsc, pisc);

  for (int l = 0; l < LAYERS; l++) {
    int is_full = (l == 4);
    const float* cv = is_full ? cos_f : cos_s;
    const float* sv = is_full ? sin_f : sin_s;

    // x = rms(h, ln_in[l])
    rms_k<<<1, 256, 0, stream>>>(wx, wh, ln_in + l * HDIM, HDIM);

    // q,k,v projections
    gemv(wx, Wq + (size_t)l * HDIM * NH * HD,  wq,  HDIM, NH * HD);
    gemv(wx, Wk + (size_t)l * HDIM * NKV * HD, wkv, HDIM, NKV * HD);
    gemv(wx, Wv + (size_t)l * HDIM * NKV * HD, wvv, HDIM, NKV * HD);

    // per-head norm + rope
    qkvnorm_k<<<NH + 2 * NKV, HD, 0, stream>>>(wq, wkv, wvv,
        qn + l * HD, kn + l * HD, cv, sv);

    // cache update (writes directly to d_out; attention reads it back)
    const float* Kcur; const float* Vcur; int Lbuf, nkeys;
    if (is_full) {
      fullupd_k<<<(NKV * CTX * HD + 255) / 256, 256, 0, stream>>>(
          Kf, Vf, Kfin, Vfin, wkv, wvv);
      Kcur = Kf; Vcur = Vf; Lbuf = CTX; nkeys = POSI + 1;
    } else {
      int si = SI[l];
      float* Ko = Ks + (size_t)si * NKV * WIN * HD;
      float* Vo = Vs + (size_t)si * NKV * WIN * HD;
      slide_k<<<(NKV * WIN * HD + 255) / 256, 256, 0, stream>>>(
          Ko, Vo,
          Ksin + (size_t)si * NKV * WIN * HD,
          Vsin + (size_t)si * NKV * WIN * HD, wkv, wvv);
      Kcur = Ko; Vcur = Vo; Lbuf = WIN; nkeys = WIN;
    }

    // attention (WMMA) -> ao head-major (== transpose+reshape in ref)
    attn_k<<<NKV, 256, 0, stream>>>(wq, Kcur, Vcur, wao, Lbuf, nkeys);

    // h += rms(ao @ Wo, ln_post_attn)
    gemv(wao, Wo + (size_t)l * NH * HD * HDIM, wy, NH * HD, HDIM);
    addrms_k<<<1, 256, 0, stream>>>(wh, wy, ln_pa + l * HDIM, HDIM, nullptr, 0);

    // MLP
    rms_k<<<1, 256, 0, stream>>>(wx, wh, ln_pf + l * HDIM, HDIM);
    gemv(wx, Wgate + (size_t)l * HDIM * DFF, wg, HDIM, DFF);
    gemv(wx, Wup   + (size_t)l * HDIM * DFF, wu, HDIM, DFF);
    glu_k<<<(DFF + 255) / 256, 256, 0, stream>>>(wff, wg, wu, DFF);
    gemv(wff, Wdown + (size_t)l * DFF * HDIM, wy, DFF, HDIM);
    addrms_k<<<1, 256, 0, stream>>>(wh, wy, ln_ff + l * HDIM, HDIM, nullptr, 0);

    // per-layer embedding block (uses h directly), then layer_scalar
    gemv(wh, Wplg + (size_t)l * HDIM * PLD, wpg, HDIM, PLD);
    plgate_k<<<1, 256, 0, stream>>>(wpg, wcomb + l * PLD, PLD);
    gemv(wpg, Wplp + (size_t)l * PLD * HDIM, wy, PLD, HDIM);
    addrms_k<<<1, 256, 0, stream>>>(wh, wy, ln_pl + l * HDIM, HDIM, lsc, l);
  }

  // final hidden
  copy_k<<<8, 256, 0, stream>>>(out, wh, HDIM);
}